// BinauralAdaptiveGammatoneFB_SingleController_49701361549329
// MI455X (gfx1250) — compile-verified
//
#include <hip/hip_runtime.h>
#include <math.h>
#include <stdint.h>

// ---------------------------------------------------------------------------
// BinauralAdaptiveGammatoneFB single-controller forward for MI455X (gfx1250)
//
//   k_consts  : ERB constants fc / Q0 / dQ (100 bands)
//   k_twiddle : 844x1040 f32 DFT matrix win[k]*[cos | -sin] (Hann folded in)
//   k_dft     : f32 WMMA GEMM (9728 frames x 842) @ (842 x 1026) -> XL/XR
//               A tile staged via GLOBAL_LOAD_ASYNC_TO_LDS_B32 (ASYNCcnt)
//   k_mag     : |X| magnitudes into workspace
//   k_scan    : ONE persistent kernel for the whole 19-step scan
//               (batches independent -> LDS-resident state, no grid sync)
// ---------------------------------------------------------------------------

#define B_      256
#define T_      19
#define FF_     513
#define NB_     100
#define WIN_    842
#define HOP_    842
#define FS_     16000
#define FRAMES_ 9728            // 2 * B * T
#define MTILES_ 608             // FRAMES/16
#define NTILES_ 65              // ceil(1026/16)
#define TWK_    844             // K padded to multiple of 4
#define TWN_    1040            // N padded to multiple of 16
#define ASTR_   849             // LDS row stride (conflict-free A-fragment reads)
#define CPLX_   4990464         // total complex values (both channels)
#define BPB_    2               // batches per scan block
// d_out float offsets (tuple: YLs, YRs, Qs, XL, XR)
#define OUT_YL  0
#define OUT_YR  486400
#define OUT_QS  972800
#define OUT_XL  1459200
#define OUT_XR  6449664

typedef float v2f __attribute__((ext_vector_type(2)));
typedef float v8f __attribute__((ext_vector_type(8)));

__device__ __forceinline__ float sigm(float x) { return 1.0f / (1.0f + expf(-x)); }

// --------------------------- constants -------------------------------------
__device__ __forceinline__ float erb_rate(float f) {
    return 21.4f * log10f(4.37e-3f * f + 1.0f);
}

__global__ void k_consts(float* fc, float* q0, float* dq) {
    int n = threadIdx.x;
    if (n >= NB_) return;
    float Emin = erb_rate(50.0f);
    float Emax = erb_rate(7200.0f);            // FS/2 * 0.9
    float E    = Emin + (Emax - Emin) * ((float)n / 99.0f);
    float f    = (exp10f(E / 21.4f) - 1.0f) * (1000.0f / 4.37f);
    float erb  = 24.7f * (4.37e-3f * f + 1.0f);
    fc[n] = f;
    q0[n] = f / (1.019f * erb);
    float En = (E - Emin) / (Emax - Emin + 1e-12f);
    dq[n] = 1.0f + En;                         // clip(2*(0.5+0.5*En), 0.001) == 1+En
}

// --------------------------- twiddle table (window folded in) ---------------
__global__ void k_twiddle(float* tw) {
    int i = blockIdx.x * blockDim.x + threadIdx.x;
    if (i >= TWK_ * TWN_) return;
    int k = i / TWN_;
    int n = i - k * TWN_;
    float v = 0.0f;
    const float PHI = 6.28318530717958647692f / 1024.0f;
    if (k < WIN_) {
        float win = 0.5f - 0.5f * cosf((6.28318530717958647692f / 842.0f) * (float)k);
        if (n < FF_) {                         // real: win*cos(2*pi*f*k/1024)
            int p = (n * k) & 1023;
            v = win * cosf((float)p * PHI);
        } else if (n < 2 * FF_) {              // imag: -win*sin(2*pi*f*k/1024)
            int p = ((n - FF_) * k) & 1023;
            v = -win * sinf((float)p * PHI);
        }
    }
    tw[i] = v;
}

// --------------------------- DFT via f32 WMMA -------------------------------
// C[9728 x 1026] = A[9728 x 842] * TW[842 x 1026], f32, V_WMMA_F32_16X16X4_F32
__global__ __launch_bounds__(128) void k_dft(const float* __restrict__ wavL,
                                             const float* __restrict__ wavR,
                                             const float* __restrict__ tw,
                                             float* __restrict__ out) {
    __shared__ float As[16 * ASTR_];           // 54.3 KB, padded stride
    int tid = threadIdx.x;
    int mt  = blockIdx.x;                      // 0..607

    // ---- stage one 16-frame x 842-sample tile: async global -> LDS ----
    // LDS address comes from the lane's VGPR; low 32 bits of a generic LDS pointer
    // are the LDS byte offset (aperture truncation).
    for (int i = tid; i < 16 * WIN_; i += 128) {
        int row = i / WIN_;
        int k   = i - row * WIN_;
        int m   = mt * 16 + row;
        int ch  = m / (B_ * T_);
        int rem = m - ch * (B_ * T_);          // b*19 + t
        int b   = rem / T_;
        int t   = rem - b * T_;
        const float* src = (ch ? wavR : wavL) + b * FS_ + t * HOP_ + k;
        unsigned           lds = (unsigned)(uintptr_t)&As[row * ASTR_ + k];
        unsigned long long ga  = (unsigned long long)(uintptr_t)src;
        asm volatile("global_load_async_to_lds_b32 %0, %1, off"
                     :: "v"(lds), "v"(ga) : "memory");
    }
    // zero-fill the pad columns (k in [842, 849))
    for (int i = tid; i < 16 * (ASTR_ - WIN_); i += 128) {
        int row = i / (ASTR_ - WIN_);
        int k   = WIN_ + (i - row * (ASTR_ - WIN_));
        As[row * ASTR_ + k] = 0.0f;
    }
    asm volatile("s_wait_asynccnt 0" ::: "memory");
    __syncthreads();

    int wid  = tid >> 5;
    int lane = tid & 31;
    int rowA = lane & 15;                      // A: M = lane%16
    int kg   = (lane >> 4) * 2;                // A/B: K = kb + vgpr + 2*(lane/16)
    int colL = lane & 15;                      // B/D: N = lane%16
    int hi   = lane >> 4;

    for (int nt = wid; nt < NTILES_; nt += 4) {
        v8f acc = {};
        int ncol = nt * 16 + colL;
        const float* ar  = As + rowA * ASTR_ + kg;
        const float* twp = tw + (size_t)kg * TWN_ + ncol;
        for (int kb = 0; kb < TWK_; kb += 4) {
            v2f a, b;
            a.x = ar[kb];
            a.y = ar[kb + 1];
            b.x = twp[(size_t)kb * TWN_];
            b.y = twp[(size_t)(kb + 1) * TWN_];
            acc = __builtin_amdgcn_wmma_f32_16x16x4_f32(
                false, a, false, b, (short)0, acc, false, false);
        }
        if (ncol < 2 * FF_) {                  // skip N padding columns
            bool im = (ncol >= FF_);
            int  f  = im ? (ncol - FF_) : ncol;
            for (int r = 0; r < 8; ++r) {      // D: M = r + 8*(lane/16)
                int m   = mt * 16 + r + 8 * hi;
                int ch  = m / (B_ * T_);
                int rem = m - ch * (B_ * T_);  // b*19 + t
                size_t base = ch ? (size_t)OUT_XR : (size_t)OUT_XL;
                out[base + (size_t)rem * (2 * FF_) + 2 * f + (im ? 1 : 0)] = acc[r];
            }
        }
    }
}

// --------------------------- magnitudes ------------------------------------
__global__ void k_mag(const float* __restrict__ out, float* __restrict__ mag) {
    int i = blockIdx.x * blockDim.x + threadIdx.x;
    if (i >= CPLX_) return;
    float re = out[OUT_XL + 2 * (size_t)i];
    float im = out[OUT_XL + 2 * (size_t)i + 1];
    mag[i] = sqrtf(re * re + im * im);
}

// --------------------------- LayerNorm + SiLU (per-batch, in-block) ---------
__device__ __forceinline__ float ln_silu2(float x, float g, float be,
                                          float* red, int bb, int j, bool act) {
    if (act) red[bb * 128 + j] = x;
    __syncthreads();
    for (int s = 64; s; s >>= 1) {
        if (act && j < s) red[bb * 128 + j] += red[bb * 128 + j + s];
        __syncthreads();
    }
    float mean = red[bb * 128] * (1.0f / 128.0f);
    __syncthreads();
    float d = x - mean;
    if (act) red[bb * 128 + j] = d * d;
    __syncthreads();
    for (int s = 64; s; s >>= 1) {
        if (act && j < s) red[bb * 128 + j] += red[bb * 128 + j + s];
        __syncthreads();
    }
    float var = red[bb * 128] * (1.0f / 128.0f);
    __syncthreads();
    float y = d * rsqrtf(var + 1e-5f) * g + be;
    return y * sigm(y);                        // SiLU
}

// --------------------------- fused 19-step scan ------------------------------
// One block per BPB_ batches; Q/h/mem state lives in LDS across all steps.
__global__ __launch_bounds__(512) void k_scan(
    const float* __restrict__ mag,  const float* __restrict__ fcc,
    const float* __restrict__ q0c,  const float* __restrict__ dqc,
    const float* __restrict__ w_ih, const float* __restrict__ w_hh,
    const float* __restrict__ b_ih, const float* __restrict__ b_hh,
    const float* __restrict__ w1, const float* __restrict__ b1,
    const float* __restrict__ g1, const float* __restrict__ be1,
    const float* __restrict__ w2, const float* __restrict__ b2,
    const float* __restrict__ g2, const float* __restrict__ be2,
    const float* __restrict__ w3, const float* __restrict__ b3,
    float* __restrict__ out) {
    __shared__ float featS[BPB_ * 400];
    __shared__ float hS[BPB_ * 128];
    __shared__ float giS[BPB_ * 384], ghS[BPB_ * 384];
    __shared__ float QstS[BPB_ * NB_], mLS[BPB_ * NB_], mRS[BPB_ * NB_];
    __shared__ float aS[BPB_ * 128], a2S[BPB_ * 128], red[BPB_ * 128];

    int tid = threadIdx.x;
    int b0  = blockIdx.x * BPB_;

    // state init (every launch; deterministic)
    for (int i = tid; i < BPB_ * 128; i += 512) hS[i] = 0.0f;
    for (int i = tid; i < BPB_ * NB_; i += 512) {
        QstS[i] = q0c[i % NB_];
        mLS[i]  = 0.0f;
        mRS[i]  = 0.0f;
    }
    __syncthreads();

    int  wv   = tid >> 5;
    int  lane = tid & 31;
    bool act  = tid < BPB_ * 128;
    int  bb   = (tid >> 7) & (BPB_ - 1);
    int  j    = tid & 127;

    for (int t = 0; t < T_; ++t) {
        // ---- phase 1: adaptive gammatone filterbank (one wave per band) ----
        for (int task = wv; task < BPB_ * NB_; task += 16) {
            int tb = task / NB_;
            int n  = task - tb * NB_;
            float Q   = QstS[task];
            float fcn = fcc[n];
            float ibw = 1.0f / (fcn / (Q + 1e-8f) + 1e-8f);
            const float* xl = mag + (size_t)((b0 + tb) * T_ + t) * FF_;
            const float* xr = xl + (size_t)(B_ * T_) * FF_;
            float sw = 0.0f, sl = 0.0f, sr = 0.0f;
            for (int f = lane; f < FF_; f += 32) {
                float d = ((float)f * 15.625f - fcn) * ibw;  // F_FFT[f]=f*8000/512
                float w = expf(-0.5f * d * d);
                sw += w; sl += w * xl[f]; sr += w * xr[f];
            }
            for (int o = 16; o; o >>= 1) {
                sw += __shfl_xor(sw, o, 32);
                sl += __shfl_xor(sl, o, 32);
                sr += __shfl_xor(sr, o, 32);
            }
            if (lane == 0) {
                float inv = 1.0f / (sw + 1e-8f);
                float yl = sl * inv, yr = sr * inv;
                if (!isfinite(yl)) yl = 0.0f;
                if (!isfinite(yr)) yr = 0.0f;
                size_t oi = (size_t)((b0 + tb) * T_ + t) * NB_ + n;
                out[OUT_YL + oi] = yl;
                out[OUT_YR + oi] = yr;
                out[OUT_QS + oi] = Q;          // Qrec = Q before update
                float cl = log1pf(fmaxf(yl, 0.0f));
                float cr = log1pf(fmaxf(yr, 0.0f));
                float ml = mLS[task], mr = mRS[task];
                float* fb = featS + tb * 400;
                fb[n]       = cl;
                fb[100 + n] = ml;
                fb[200 + n] = cr;
                fb[300 + n] = mr;
                mLS[task] = 0.8f * ml + 0.2f * cl;    // BETA = 0.8
                mRS[task] = 0.8f * mr + 0.2f * cr;
            }
        }
        __syncthreads();

        // ---- phase 2: GRU GEMMs (weight row reused across BPB_ batches) ----
        if (tid < 384) {
            float ai[BPB_], ah[BPB_];
            for (int q = 0; q < BPB_; ++q) ai[q] = b_ih[tid];
            const float* wi = w_ih + (size_t)tid * 400;
            for (int k = 0; k < 400; ++k) {
                float wvv = wi[k];
                for (int q = 0; q < BPB_; ++q) ai[q] += featS[q * 400 + k] * wvv;
            }
            for (int q = 0; q < BPB_; ++q) giS[q * 384 + tid] = ai[q];
            for (int q = 0; q < BPB_; ++q) ah[q] = b_hh[tid];
            const float* wh = w_hh + (size_t)tid * 128;
            for (int k = 0; k < 128; ++k) {
                float wvv = wh[k];
                for (int q = 0; q < BPB_; ++q) ah[q] += hS[q * 128 + k] * wvv;
            }
            for (int q = 0; q < BPB_; ++q) ghS[q * 384 + tid] = ah[q];
        }
        __syncthreads();

        // ---- phase 3: GRU gates ----
        if (act) {
            const float* gi = giS + bb * 384;
            const float* gh = ghS + bb * 384;
            float r  = sigm(gi[j] + gh[j]);
            float z  = sigm(gi[128 + j] + gh[128 + j]);
            float nn = tanhf(gi[256 + j] + r * gh[256 + j]);
            hS[bb * 128 + j] = (1.0f - z) * nn + z * hS[bb * 128 + j];
        }
        __syncthreads();

        // ---- phase 4: MLP layer 1 ----
        float acc = 0.0f;
        if (act) {
            acc = b1[j];
            const float* wr = w1 + (size_t)j * 128;
            const float* hp = hS + bb * 128;
            for (int k = 0; k < 128; ++k) acc += hp[k] * wr[k];
        }
        acc = ln_silu2(acc, g1[j], be1[j], red, bb, j, act);
        if (act) aS[bb * 128 + j] = acc;
        __syncthreads();

        // ---- phase 5: MLP layer 2 ----
        if (act) {
            acc = b2[j];
            const float* wr = w2 + (size_t)j * 128;
            const float* ap = aS + bb * 128;
            for (int k = 0; k < 128; ++k) acc += ap[k] * wr[k];
        }
        acc = ln_silu2(acc, g2[j], be2[j], red, bb, j, act);
        if (act) a2S[bb * 128 + j] = acc;
        __syncthreads();

        // ---- phase 6: head + Q update ----
        if (tid < BPB_ * NB_) {
            int tb = tid / NB_;
            int n  = tid - tb * NB_;
            float a3 = b3[n];
            const float* w3r = w3 + (size_t)n * 128;
            const float* ap  = a2S + tb * 128;
            for (int k = 0; k < 128; ++k) a3 += ap[k] * w3r[k];
            float delta = tanhf(a3);
            float qt = q0c[n] + delta * dqc[n];       // absolute deltaQ mode
            float qn = 0.8f * QstS[tid] + 0.2f * qt;  // ALPHA = 0.2
            QstS[tid] = fminf(fmaxf(qn, 0.05f), 30.0f);
        }
        __syncthreads();
    }
}

// --------------------------- launch ----------------------------------------
extern "C" void kernel_launch(void* const* d_in, const int* in_sizes, int n_in,
                              void* d_out, int out_size, void* d_ws, size_t ws_size,
                              hipStream_t stream) {
    (void)in_sizes; (void)n_in; (void)out_size; (void)ws_size;
    const float* wavL = (const float*)d_in[0];
    const float* wavR = (const float*)d_in[1];
    const float* w_ih = (const float*)d_in[2];
    const float* w_hh = (const float*)d_in[3];
    const float* b_ih = (const float*)d_in[4];
    const float* b_hh = (const float*)d_in[5];
    const float* w1   = (const float*)d_in[6];
    const float* b1   = (const float*)d_in[7];
    const float* g1   = (const float*)d_in[8];
    const float* be1  = (const float*)d_in[9];
    const float* w2   = (const float*)d_in[10];
    const float* b2   = (const float*)d_in[11];
    const float* g2   = (const float*)d_in[12];
    const float* be2  = (const float*)d_in[13];
    const float* w3   = (const float*)d_in[14];
    const float* b3   = (const float*)d_in[15];
    float* out = (float*)d_out;

    // workspace layout (floats)
    float* W   = (float*)d_ws;
    float* TW  = W;                            // 844*1040 = 877,760
    float* FCC = TW + (size_t)TWK_ * TWN_;     // 128
    float* Q0C = FCC + 128;                    // 128
    float* DQC = Q0C + 128;                    // 128
    float* MAG = DQC + 128;                    // 4,990,464

    k_consts<<<1, 128, 0, stream>>>(FCC, Q0C, DQC);
    {
        int n = TWK_ * TWN_;
        k_twiddle<<<(n + 255) / 256, 256, 0, stream>>>(TW);
    }
    k_dft<<<MTILES_, 128, 0, stream>>>(wavL, wavR, TW, out);
    k_mag<<<(CPLX_ + 255) / 256, 256, 0, stream>>>(out, MAG);
    k_scan<<<B_ / BPB_, 512, 0, stream>>>(MAG, FCC, Q0C, DQC,
                                          w_ih, w_hh, b_ih, b_hh,
                                          w1, b1, g1, be1, w2, b2, g2, be2,
                                          w3, b3, out);
}